// MMLorentzKG_86337432584674
// MI455X (gfx1250) — compile-verified
//
#include <hip/hip_runtime.h>
#include <math.h>

// Problem constants (from reference)
#define RANK      64
#define ENT_COMPS 5
#define D         320          // ENT_COMPS*RANK
#define KDIM      640          // 2*D : GEMM reduction dim; also entity row stride
#define N_ENT     80000
#define B_SZ      2000
#define SCALE_F   0.125f       // 1/sqrt(RANK)
#define EPS_F     1e-9f

#define M_TILES   125          // 2000 / 16, exact
#define LDSTRIDE  644          // 640 + 4 pad: conflict-free b64 reads, 16B-aligned rows

typedef float v2f __attribute__((ext_vector_type(2)));
typedef float v8f __attribute__((ext_vector_type(8)));

// ---------------------------------------------------------------------------
// Kernel 1: per-query prep. 1 block per batch element, 64 threads (1/rank).
// Produces P = [P0|P1] (B,640) into workspace, and reg0..reg4 into d_out.
// ---------------------------------------------------------------------------
__global__ __launch_bounds__(64) void prep_kernel(
    const int*   __restrict__ queries,   // (B,3)
    const float* __restrict__ ent,       // (N_ENT,128,5) -> rows of 640 floats
    const float* __restrict__ rrot,      // (N_REL,64,4)
    const float* __restrict__ rboost,    // (N_REL,64,4)
    const float* __restrict__ atten,     // (N_REL,1,320)
    const float* __restrict__ aux,       // (N_REL,2,320)
    float* __restrict__ P,               // (B,640) workspace
    float* __restrict__ reg0,            // (B,320)
    float* __restrict__ reg1,            // (B,320)
    float* __restrict__ reg2,            // (B,320)
    float* __restrict__ reg3,            // (B,64)
    float* __restrict__ reg4)            // (B,64)
{
    const int b = blockIdx.x;
    const int j = threadIdx.x;                    // rank index 0..63

    const int h = queries[b * 3 + 0];
    const int r = queries[b * 3 + 1];
    const int t = queries[b * 3 + 2];

    const float* e_h = ent + (size_t)h * KDIM;
    const float* e_t = ent + (size_t)t * KDIM;

    // lhs_re[b, j, 0..4]
    float x[5];
#pragma unroll
    for (int c = 0; c < 5; ++c) x[c] = e_h[j * 5 + c];

    // ---- lorentz_rotate ----
    const float* q4 = rrot + ((size_t)r * RANK + j) * 4;
    float qw = q4[0], qx = q4[1], qy = q4[2], qz = q4[3];
    const float qn2 = qw * qw + qx * qx + qy * qy + qz * qz;   // for reg3
    const float qi  = rsqrtf(qn2 + EPS_F);
    qw *= qi; qx *= qi; qy *= qi; qz *= qi;
    const float pw = x[1], px = x[2], py = x[3], pz = x[4];
    float rot[5];
    rot[0] = x[0];
    rot[1] = qw * pw - qx * px - qy * py - qz * pz;
    rot[2] = qw * px + qx * pw + qy * pz - qz * py;
    rot[3] = qw * py - qx * pz + qy * pw + qz * px;
    rot[4] = qw * pz + qx * py - qy * px + qz * pw;

    // ---- lorentz_boost ----
    const float* v4 = rboost + ((size_t)r * RANK + j) * 4;
    const float v0 = v4[0], v1 = v4[1], v2 = v4[2], v3 = v4[3];
    const float vn2raw = v0 * v0 + v1 * v1 + v2 * v2 + v3 * v3; // for reg4
    const float vn2 = fminf(fmaxf(vn2raw, 0.0f), 0.999f);
    const float gamma = rsqrtf(1.0f - vn2);
    const float vs = v0 * x[1] + v1 * x[2] + v2 * x[3] + v3 * x[4];
    float boo[5];
    boo[0] = gamma * (x[0] + vs);
    const float coef = gamma * x[0] + (gamma - 1.0f) * vs / (vn2 + EPS_F);
    boo[1] = x[1] + coef * v0;
    boo[2] = x[2] + coef * v1;
    boo[3] = x[3] + coef * v2;
    boo[4] = x[4] + coef * v3;

    // ---- attention logits: dot(atten, rot) and dot(atten, boo) over 320 ----
    const float* att = atten + (size_t)r * D;
    float p0 = 0.0f, p1 = 0.0f;
#pragma unroll
    for (int c = 0; c < 5; ++c) {
        const float a = att[j * 5 + c];
        p0 += a * rot[c];
        p1 += a * boo[c];
    }
    __shared__ float s0[64];
    __shared__ float s1[64];
    s0[j] = p0; s1[j] = p1;
    __syncthreads();
#pragma unroll
    for (int st = 32; st > 0; st >>= 1) {
        if (j < st) { s0[j] += s0[j + st]; s1[j] += s1[j + st]; }
        __syncthreads();
    }
    const float l0 = s0[0] * SCALE_F;
    const float l1 = s1[0] * SCALE_F;
    const float mx = fmaxf(l0, l1);
    const float e0 = expf(l0 - mx);
    const float e1 = expf(l1 - mx);
    const float inv = 1.0f / (e0 + e1);
    const float w0 = e0 * inv, w1 = e1 * inv;

    // ---- form P rows + regs ----
    const float* aux0 = aux + (size_t)r * (2 * D);
    const float* aux1 = aux0 + D;
    float* Prow = P + (size_t)b * KDIM;
#pragma unroll
    for (int c = 0; c < 5; ++c) {
        const int d = j * 5 + c;
        const float lhs = rot[c] * w0 + boo[c] * w1;
        const float li  = e_h[D + d];
        const float a0  = aux0[d], a1 = aux1[d];
        Prow[d]     = lhs * a0 - li * a1;
        Prow[D + d] = lhs * a1 + li * a0;
        reg0[(size_t)b * D + d] = sqrtf(x[c] * x[c] + li * li);
        reg1[(size_t)b * D + d] = sqrtf(a0 * a0 + a1 * a1);
        const float tre = e_t[d], tim = e_t[D + d];
        reg2[(size_t)b * D + d] = sqrtf(tre * tre + tim * tim) * (1.0f / 3.0f);
    }
    reg3[(size_t)b * RANK + j] = sqrtf(qn2);
    reg4[(size_t)b * RANK + j] = sqrtf(vn2raw);
}

// ---------------------------------------------------------------------------
// Kernel 2: scores = P(2000x640) @ E(80000x640)^T via V_WMMA_F32_16X16X4_F32.
// Each block: 32 entity columns staged in LDS, loops all 125 M-tiles.
// Each wave: one 16-row stripe, TWO 16x16 tiles sharing the A fragment.
// ---------------------------------------------------------------------------
__global__ __launch_bounds__(128) void gemm_kernel(
    const float* __restrict__ P,       // (2000,640)
    const float* __restrict__ ent,     // (80000,640) = [to_re|to_im] per row
    float* __restrict__ scores)        // (2000,80000)
{
    __shared__ float ldsE[32 * LDSTRIDE];   // ~82.4 KB

    const int n0  = blockIdx.x * 32;
    const int tid = threadIdx.x;

    // Cooperative load of 32 entity rows (640 f32 each) as float4.
    for (int idx = tid; idx < 32 * (KDIM / 4); idx += 128) {
        const int row = idx / (KDIM / 4);
        const int cv  = idx % (KDIM / 4);
        const float4 v = *((const float4*)(ent + (size_t)(n0 + row) * KDIM) + cv);
        *((float4*)(&ldsE[row * LDSTRIDE]) + cv) = v;
    }
    __syncthreads();

    const int wave = tid >> 5;
    const int lane = tid & 31;
    const int half = lane >> 4;     // 0: K=0,1 / rows v ; 1: K=2,3 / rows v+8
    const int l16  = lane & 15;

    const float* b0p = &ldsE[(l16)      * LDSTRIDE + 2 * half];
    const float* b1p = &ldsE[(16 + l16) * LDSTRIDE + 2 * half];

    for (int mt = wave; mt < M_TILES; mt += 4) {
        const int m0 = mt * 16;
        const float* aptr = P + (size_t)(m0 + l16) * KDIM + 2 * half;

        v8f c0 = {};
        v8f c1 = {};
#pragma unroll 8
        for (int k = 0; k < KDIM; k += 4) {
            const v2f a  = *(const v2f*)(aptr + k);
            const v2f b0 = *(const v2f*)(b0p + k);
            const v2f b1 = *(const v2f*)(b1p + k);
            c0 = __builtin_amdgcn_wmma_f32_16x16x4_f32(
                     false, a, false, b0, (short)0, c0, false, false);
            c1 = __builtin_amdgcn_wmma_f32_16x16x4_f32(
                     false, a, false, b1, (short)0, c1, false, false);
        }

        // D layout: VGPR v -> row m0 + v + 8*half, col n0 + l16 (+16 for c1)
        float* out = scores + (size_t)(m0 + half * 8) * N_ENT + n0 + l16;
#pragma unroll
        for (int v = 0; v < 8; ++v) {
            out[(size_t)v * N_ENT]      = c0[v];
            out[(size_t)v * N_ENT + 16] = c1[v];
        }
    }
}

// ---------------------------------------------------------------------------
extern "C" void kernel_launch(void* const* d_in, const int* in_sizes, int n_in,
                              void* d_out, int out_size, void* d_ws, size_t ws_size,
                              hipStream_t stream) {
    const int*   queries = (const int*)  d_in[0];
    const float* ent     = (const float*)d_in[1];
    const float* rrot    = (const float*)d_in[2];
    const float* rboost  = (const float*)d_in[3];
    const float* atten   = (const float*)d_in[4];
    const float* aux     = (const float*)d_in[5];

    float* out    = (float*)d_out;
    float* scores = out;                                   // (B, N_ENT)
    float* reg0   = out + (size_t)B_SZ * N_ENT;            // (B, 320)
    float* reg1   = reg0 + (size_t)B_SZ * D;               // (B, 320)
    float* reg2   = reg1 + (size_t)B_SZ * D;               // (B, 320)
    float* reg3   = reg2 + (size_t)B_SZ * D;               // (B, 64)
    float* reg4   = reg3 + (size_t)B_SZ * RANK;            // (B, 64)

    float* P = (float*)d_ws;   // needs B_SZ*640*4 = 5.12 MB of workspace

    prep_kernel<<<B_SZ, 64, 0, stream>>>(queries, ent, rrot, rboost, atten, aux,
                                         P, reg0, reg1, reg2, reg3, reg4);
    gemm_kernel<<<N_ENT / 32, 128, 0, stream>>>(P, ent, scores);
}